// PGLoss_11991548690975
// MI455X (gfx1250) — compile-verified
//
#include <hip/hip_runtime.h>

// Problem constants (from reference): BATCH=4096, VOCAB=50257, SEQ=20
#define PG_BATCH 4096
#define PG_VOCAB 50257
#define PG_SEQ   20
#define PG_TOTAL (PG_BATCH * PG_SEQ)   // 81920 (i,j) pairs

#define NBLK 256
#define NTHR 256

typedef __attribute__((ext_vector_type(2))) float v2f;
typedef __attribute__((ext_vector_type(8))) float v8f;

// Reduce 256 f32 values in LDS to a single sum, executed by all 32 lanes of
// wave 0 (EXEC all-ones, as WMMA requires). Uses V_WMMA_F32_16X16X4_F32 with
// A = ones(16x4):  D[m][n] = sum_k B[k][n] + C[m][n].
// Each issue consumes 64 distinct LDS values into the two B VGPRs; after 4
// accumulating issues, row 0 of D (c[0] across lanes 0..15) holds 16 column
// sums that together cover all 256 inputs exactly once. A 4-step shfl_xor
// butterfly over those 16 lanes produces the total (exact f32 arithmetic).
__device__ __forceinline__ float block256_wmma_reduce(const float* lds) {
  const int lane = threadIdx.x & 31;
  v2f a;  a.x = 1.0f; a.y = 1.0f;        // A = all ones (16x4)
  v8f c = {};                            // C/D accumulator (16x16 f32)
#pragma unroll
  for (int it = 0; it < 4; ++it) {
    v2f b;
    b.x = lds[it * 64 + lane];           // fills B VGPR0 (32 values)
    b.y = lds[it * 64 + 32 + lane];      // fills B VGPR1 (32 values)
    c = __builtin_amdgcn_wmma_f32_16x16x4_f32(
        /*neg_a=*/false, a, /*neg_b=*/false, b,
        /*c_mod=*/(short)0, c, /*reuse_a=*/false, /*reuse_b=*/false);
  }
  // c[0]: lanes 0..15 = D[0][0..15] (lanes 16..31 replicate row 8 == row 0).
  float r = c[0];
#pragma unroll
  for (int off = 8; off >= 1; off >>= 1)
    r += __shfl_xor(r, off, 32);
  return r;  // valid in lane 0
}

// Kernel 1: grid-stride gather + multiply-accumulate, one partial per block.
__global__ void __launch_bounds__(NTHR)
pg_partial_kernel(const float* __restrict__ pred,
                  const int*   __restrict__ target,
                  const float* __restrict__ reward,
                  float*       __restrict__ partial) {
  __shared__ float lds[NTHR];
  float acc = 0.0f;
  const int stride = NTHR * gridDim.x;
  for (int idx = blockIdx.x * NTHR + threadIdx.x; idx < PG_TOTAL; idx += stride) {
    const int i = idx / PG_SEQ;                     // row in pred
    const int t = target[idx];                      // column (int32 index)
    acc = fmaf(pred[(long long)i * PG_VOCAB + t], reward[idx], acc);
  }
  lds[threadIdx.x] = acc;
  __syncthreads();
  if (threadIdx.x < 32) {                           // wave 0 only, EXEC = ~0
    const float s = block256_wmma_reduce(lds);
    if (threadIdx.x == 0) partial[blockIdx.x] = s;
  }
}

// Kernel 2: single block reduces NBLK partials, writes final scalar.
__global__ void __launch_bounds__(NTHR)
pg_final_kernel(const float* __restrict__ partial, float* __restrict__ out) {
  __shared__ float lds[NTHR];
  lds[threadIdx.x] = partial[threadIdx.x];
  __syncthreads();
  if (threadIdx.x < 32) {
    const float s = block256_wmma_reduce(lds);
    if (threadIdx.x == 0) out[0] = -s * (1.0f / (float)PG_BATCH);
  }
}

extern "C" void kernel_launch(void* const* d_in, const int* in_sizes, int n_in,
                              void* d_out, int out_size, void* d_ws, size_t ws_size,
                              hipStream_t stream) {
  (void)in_sizes; (void)n_in; (void)out_size; (void)ws_size;
  const float* pred   = (const float*)d_in[0];   // (4096, 50257) f32
  const int*   target = (const int*)d_in[1];     // (4096, 20) int indices
  const float* reward = (const float*)d_in[2];   // (4096, 20) f32
  float*       out    = (float*)d_out;           // scalar f32
  float*       partial = (float*)d_ws;           // NBLK floats of scratch

  pg_partial_kernel<<<NBLK, NTHR, 0, stream>>>(pred, target, reward, partial);
  pg_final_kernel<<<1, NTHR, 0, stream>>>(partial, out);
}